// FasterRCNN_Network_18270790877598
// MI455X (gfx1250) — compile-verified
//
#include <hip/hip_runtime.h>
#include <hip/hip_bf16.h>
#include <stdint.h>

typedef float v2f __attribute__((ext_vector_type(2)));
typedef float v8f __attribute__((ext_vector_type(8)));
typedef unsigned long long u64;
typedef unsigned int u32;

#define HF 128
#define WF 128
#define NAA 9
#define NANCH (HF * WF * NAA)      // 147456
#define PRE 6000
#define POST 300
#define SORTN 8192
#define MASKW 94                   // ceil(6000/64)
#define NMS_T 0.7f
#define MINSZ 3.0f
#define NTILE (PRE / 16)           // 375

// ---- workspace layout (bytes), all 16B aligned ----
#define OFF_KEYS  ((size_t)0)
#define OFF_BOXES ((size_t)(OFF_KEYS + 4ull * NANCH))          // u32 keys
#define OFF_HIST  ((size_t)(OFF_BOXES + 16ull * NANCH))        // float4 boxes
#define OFF_STATE ((size_t)(OFF_HIST + 1024))                  // 256 u32 hist
#define OFF_SEL   ((size_t)(OFF_STATE + 256))                  // 64 u32 state
#define OFF_SBOX  ((size_t)(OFF_SEL + 8ull * SORTN))           // u64 pairs
#define OFF_SKEY  ((size_t)(OFF_SBOX + 16ull * PRE))           // float4 sorted boxes
#define OFF_MASK  ((size_t)(OFF_SKEY + 4ull * PRE))            // u32 sorted keys
// mask: PRE*MASKW u64  (4.5 MB);  total ws ~7.7 MB

__constant__ float c_anchors[NAA][4] = {
  {-84.f,-40.f,99.f,55.f},{-176.f,-88.f,191.f,103.f},{-360.f,-184.f,375.f,199.f},
  {-56.f,-56.f,71.f,71.f},{-120.f,-120.f,135.f,135.f},{-248.f,-248.f,263.f,263.f},
  {-36.f,-80.f,51.f,95.f},{-80.f,-168.f,95.f,183.f},{-168.f,-344.f,183.f,359.f}};

__device__ __forceinline__ u32 fkey(float f) {            // monotonic float->u32
  u32 u = __float_as_uint(f);
  return (u & 0x80000000u) ? ~u : (u | 0x80000000u);
}

// ---------------- init: zero mask / selPairs / hist / state -----------------
__global__ void init_kernel(u64* mask, u64* selPairs, u32* hist, u32* state) {
  size_t i = (size_t)blockIdx.x * blockDim.x + threadIdx.x;
  if (i < (size_t)PRE * MASKW) mask[i] = 0ull;
  if (i < SORTN) selPairs[i] = 0ull;
  if (i < 256) hist[i] = 0u;
  if (i == 0) { state[0] = 0u; state[1] = 0u; state[2] = PRE; state[3] = 0u; }
}

// ---------------- scores + bbox decode ----------------
__global__ void score_box_kernel(const float* __restrict__ cls,
                                 const float* __restrict__ bpred,
                                 const float* __restrict__ iminfo,
                                 u32* __restrict__ keys,
                                 float* __restrict__ boxes) {
  int t = blockIdx.x * blockDim.x + threadIdx.x;
  if (t >= NANCH) return;
  const int HW = HF * WF;
  int a = t % NAA;
  int p = t / NAA;            // h*WF + w
  int wwi = p % WF;
  int hh  = p / WF;

  __builtin_prefetch(&bpred[(4 * a) * HW + p], 0, 0);   // global_prefetch_b8

  float bg = cls[a * HW + p];
  float fg = cls[(NAA + a) * HW + p];
  float score = 1.0f / (1.0f + expf(bg - fg));          // softmax fg prob

  float dx = bpred[(4 * a + 0) * HW + p];
  float dy = bpred[(4 * a + 1) * HW + p];
  float dw = bpred[(4 * a + 2) * HW + p];
  float dh = bpred[(4 * a + 3) * HW + p];

  float ax1 = c_anchors[a][0] + 16.f * wwi;
  float ay1 = c_anchors[a][1] + 16.f * hh;
  float ax2 = c_anchors[a][2] + 16.f * wwi;
  float ay2 = c_anchors[a][3] + 16.f * hh;
  float wa = ax2 - ax1 + 1.f, ha = ay2 - ay1 + 1.f;
  float cx = ax1 + 0.5f * wa, cy = ay1 + 0.5f * ha;
  float pcx = dx * wa + cx, pcy = dy * ha + cy;
  float pw = expf(dw) * wa, ph = expf(dh) * ha;
  float x1 = pcx - 0.5f * pw, y1 = pcy - 0.5f * ph;
  float x2 = pcx + 0.5f * pw, y2 = pcy + 0.5f * ph;

  float imh = iminfo[0], imw = iminfo[1], ims = iminfo[2];
  x1 = fminf(fmaxf(x1, 0.f), imw - 1.f);
  y1 = fminf(fmaxf(y1, 0.f), imh - 1.f);
  x2 = fminf(fmaxf(x2, 0.f), imw - 1.f);
  y2 = fminf(fmaxf(y2, 0.f), imh - 1.f);

  float bw = x2 - x1 + 1.f, bh = y2 - y1 + 1.f;
  bool valid = (bw >= MINSZ * ims) && (bh >= MINSZ * ims);
  keys[t] = valid ? fkey(score) : fkey(-INFINITY);
  float4 b4 = make_float4(x1, y1, x2, y2);
  ((float4*)boxes)[t] = b4;
}

// ---------------- radix-select (exact 6000th-largest key) ----------------
__global__ void hist_kernel(const u32* __restrict__ keys, u32* hist,
                            const u32* __restrict__ state, int shift) {
  __shared__ u32 lh[256];
  int tid = threadIdx.x;
  if (tid < 256) lh[tid] = 0u;
  __syncthreads();
  int i = blockIdx.x * blockDim.x + tid;
  if (i < NANCH) {
    u32 k = keys[i];
    if ((k & state[1]) == state[0]) atomicAdd(&lh[(k >> shift) & 255u], 1u);
  }
  __syncthreads();
  if (tid < 256 && lh[tid]) atomicAdd(&hist[tid], lh[tid]);
}

__global__ void scan_kernel(u32* hist, u32* state, int shift) {
  if (threadIdx.x == 0) {
    u32 remK = state[2];
    u32 above = 0;
    int b = 255;
    for (; b > 0; --b) {
      if (above + hist[b] >= remK) break;
      above += hist[b];
    }
    state[0] |= ((u32)b) << shift;
    state[1] |= 0xFFu << shift;
    state[2] = remK - above;
  }
  __syncthreads();
  hist[threadIdx.x] = 0u;                 // reset for next pass (256 threads)
}

__global__ void compact_kernel(const u32* __restrict__ keys,
                               const u32* __restrict__ state,
                               u32* counter, u64* selPairs) {
  int i = blockIdx.x * blockDim.x + threadIdx.x;
  if (i >= NANCH) return;
  u32 k = keys[i];
  if (k >= state[0]) {                    // state[0] == kth-largest key
    u32 pos = atomicAdd(counter, 1u);
    if (pos < SORTN)                      // key desc, then index asc
      selPairs[pos] = ((u64)k << 32) | (u64)(0xFFFFFFFFu - (u32)i);
  }
}

// ------- single-block bitonic sort of 8192 pairs; async global->LDS stage ----
__global__ __launch_bounds__(1024)
void sort_gather_kernel(const u64* __restrict__ selPairs,
                        const float* __restrict__ boxes,
                        float* __restrict__ sortedBoxes,
                        u32* __restrict__ sortedKey) {
  __shared__ u64 sbuf[SORTN];             // 64 KB LDS
  int tid = threadIdx.x;

  // Async bulk copy: 8 u64 per thread via 4x b128 (ASYNCcnt path).
  {
    u32 ldsa = (u32)(size_t)&sbuf[tid * 8];
    u64 ga = (u64)(const void*)(selPairs + (size_t)tid * 8);
    asm volatile("global_load_async_to_lds_b128 %0, %1, off offset:0"
                 :: "v"(ldsa), "v"(ga) : "memory");
    asm volatile("global_load_async_to_lds_b128 %0, %1, off offset:16"
                 :: "v"(ldsa), "v"(ga) : "memory");
    asm volatile("global_load_async_to_lds_b128 %0, %1, off offset:32"
                 :: "v"(ldsa), "v"(ga) : "memory");
    asm volatile("global_load_async_to_lds_b128 %0, %1, off offset:48"
                 :: "v"(ldsa), "v"(ga) : "memory");
    asm volatile("s_wait_asynccnt 0x0" ::: "memory");
  }
  __syncthreads();

  // Bitonic sort, descending (pairs already encode tie-break by index).
  for (u32 ks = 2; ks <= SORTN; ks <<= 1) {
    for (u32 js = ks >> 1; js > 0; js >>= 1) {
      for (u32 i = tid; i < SORTN; i += 1024) {
        u32 ixj = i ^ js;
        if (ixj > i) {
          u64 a = sbuf[i], b = sbuf[ixj];
          bool desc = ((i & ks) == 0);
          if (desc ? (a < b) : (a > b)) { sbuf[i] = b; sbuf[ixj] = a; }
        }
      }
      __syncthreads();
    }
  }

  for (int s = tid; s < PRE; s += 1024) {
    u64 pr = sbuf[s];
    u32 key = (u32)(pr >> 32);
    u32 idx = 0xFFFFFFFFu - (u32)(pr & 0xFFFFFFFFu);
    sortedKey[s] = key;
    float4 b4 = make_float4(0.f, 0.f, 0.f, 0.f);
    if ((key & 0x80000000u) && idx < NANCH) b4 = ((const float4*)boxes)[idx];
    ((float4*)sortedBoxes)[s] = b4;
  }
}

// ---------------- tiled pairwise IoU -> suppression bitmask (WMMA) ----------
// Block = 8 waves (dim3(32,8)); each wave owns one 16x16 tile of the pair
// matrix. No shuffles: col boxes are per-lane loads, row boxes are 2-address
// broadcast loads preloaded as one clause. Division-free IoU test.
__global__ __launch_bounds__(256)
void nms_mask_kernel(const float* __restrict__ sortedBoxes, u64* __restrict__ mask) {
  int colTile = blockIdx.x;                         // 0..374
  int rowTile = blockIdx.y * 8 + threadIdx.y;       // 0..375
  if (rowTile >= NTILE || colTile < rowTile) return;  // wave-uniform exit
  int lane = threadIdx.x;

  // Per-lane column box (lane&15 pattern makes lanes 16-31 mirror 0-15).
  float4 cb = ((const float4*)sortedBoxes)[colTile * 16 + (lane & 15)];
  float areaC = (cb.z - cb.x + 1.f) * (cb.w - cb.y + 1.f);

  // Row areas for WMMA A operand (lanes 0-15 hold row areas).
  float areaR = 0.f;
  if (lane < 16) {
    float4 rb0 = ((const float4*)sortedBoxes)[rowTile * 16 + lane];
    areaR = (rb0.z - rb0.x + 1.f) * (rb0.w - rb0.y + 1.f);
  }

  // Preload the 8 row boxes this lane needs (2 distinct addresses per wave per
  // iteration; issued as one load clause so latency overlaps).
  float4 rbs[8];
#pragma unroll
  for (int r = 0; r < 8; ++r) {
    int m = (lane < 16) ? r : (r + 8);
    rbs[r] = ((const float4*)sortedBoxes)[rowTile * 16 + m];
  }

  // Rank-2 outer sum via WMMA: D[m][n] = areaR[m]*1 + 1*areaC[n].
  // A(16x4): col0=areaR, col1=1 -> lanes0-15 hold (K0,K1); lanes16-31 (K2,K3)=0
  // B(4x16): row0=1, row1=areaC -> VGPR0: K0 (lanes0-15), K2 (16-31); VGPR1: K1,K3
  v2f A; A.x = (lane < 16) ? areaR : 0.f; A.y = (lane < 16) ? 1.f : 0.f;
  v2f B; B.x = (lane < 16) ? 1.f : 0.f;   B.y = (lane < 16) ? areaC : 0.f;
  v8f C = {0.f, 0.f, 0.f, 0.f, 0.f, 0.f, 0.f, 0.f};
  C = __builtin_amdgcn_wmma_f32_16x16x4_f32(false, A, false, B, (short)0, C,
                                            false, false);
  // D layout: VGPR r -> row r (lanes 0-15) / row r+8 (lanes 16-31), col = lane%16.

  const bool diag = (colTile == rowTile);
  const int wrd = colTile >> 2;
  const int boff = (colTile & 3) * 16;
  const int gj = colTile * 16 + (lane & 15);
#pragma unroll
  for (int r = 0; r < 8; ++r) {
    int m = (lane < 16) ? r : (r + 8);
    float4 rb = rbs[r];
    float iw = fminf(rb.z, cb.z) - fmaxf(rb.x, cb.x) + 1.f;
    float ih = fminf(rb.w, cb.w) - fmaxf(rb.y, cb.y) + 1.f;
    float inter = fmaxf(iw, 0.f) * fmaxf(ih, 0.f);
    // union = areaR[m]+areaC[n]-inter >= 1 > 0, so iou>T <=> inter > T*union
    bool pred = inter > NMS_T * (C[r] - inter);
    if (diag) pred = pred && (gj > (rowTile * 16 + m));
    u32 bal = __builtin_amdgcn_ballot_w32(pred);
    // Single guard: lane 0 writes rows r (low half bits), lane 16 writes r+8.
    u32 bits = (lane < 16) ? (bal & 0xFFFFu) : (bal >> 16);
    if ((lane & 15) == 0 && bits)
      atomicOr(&mask[(size_t)(rowTile * 16 + m) * MASKW + wrd],
               ((u64)bits) << boff);
  }
}

// ---------------- serial greedy reduction + top-300 emit ----------------
__global__ __launch_bounds__(128)
void nms_reduce_out_kernel(const u32* __restrict__ sortedKey,
                           const float* __restrict__ sortedBoxes,
                           const u64* __restrict__ mask,
                           float* __restrict__ out) {
  __shared__ u64 remv[MASKW];
  __shared__ int outIdx[POST];
  __shared__ int kcount;
  __shared__ int flag;
  __shared__ int stop;
  int tid = threadIdx.x;
  if (tid < MASKW) remv[tid] = 0ull;
  if (tid == 0) { kcount = 0; stop = 0; }
  __syncthreads();

  for (int i = 0; i < PRE; ++i) {
    if (tid == 0) {
      bool alive = !((remv[i >> 6] >> (i & 63)) & 1ull) &&
                   (sortedKey[i] & 0x80000000u);
      flag = alive ? 1 : 0;
      if (alive) {
        if (kcount < POST) outIdx[kcount] = i;
        ++kcount;
        if (kcount >= POST) stop = 1;     // only first 300 kept matter
      }
    }
    __syncthreads();
    if (stop) break;
    if (flag && tid < MASKW) remv[tid] |= mask[(size_t)i * MASKW + tid];
    __syncthreads();
  }
  __syncthreads();

  int kc = kcount < POST ? kcount : POST;
  for (int s = tid; s < POST; s += 128) {
    float4 b = make_float4(0.f, 0.f, 0.f, 0.f);
    if (s < kc) b = ((const float4*)sortedBoxes)[outIdx[s]];
    out[s * 5 + 0] = 0.f;
    out[s * 5 + 1] = b.x;
    out[s * 5 + 2] = b.y;
    out[s * 5 + 3] = b.z;
    out[s * 5 + 4] = b.w;
  }
}

// ---------------------------------------------------------------------------
extern "C" void kernel_launch(void* const* d_in, const int* in_sizes, int n_in,
                              void* d_out, int out_size, void* d_ws, size_t ws_size,
                              hipStream_t stream) {
  const float* cls    = (const float*)d_in[0];   // (1,18,128,128)
  const float* bpred  = (const float*)d_in[1];   // (1,36,128,128)
  const float* iminfo = (const float*)d_in[2];   // (1,3)
  float* out = (float*)d_out;                    // (300,5) f32

  char* ws = (char*)d_ws;
  u32*   keys        = (u32*)(ws + OFF_KEYS);
  float* boxes       = (float*)(ws + OFF_BOXES);
  u32*   hist        = (u32*)(ws + OFF_HIST);
  u32*   state       = (u32*)(ws + OFF_STATE);   // [0]=prefVal [1]=prefMask [2]=remK [3]=selCount
  u64*   selPairs    = (u64*)(ws + OFF_SEL);
  float* sortedBoxes = (float*)(ws + OFF_SBOX);
  u32*   sortedKey   = (u32*)(ws + OFF_SKEY);
  u64*   mask        = (u64*)(ws + OFF_MASK);

  int initBlocks = (int)(((size_t)PRE * MASKW + 255) / 256);
  init_kernel<<<initBlocks, 256, 0, stream>>>(mask, selPairs, hist, state);

  score_box_kernel<<<(NANCH + 255) / 256, 256, 0, stream>>>(cls, bpred, iminfo,
                                                            keys, boxes);
  for (int p = 3; p >= 0; --p) {
    hist_kernel<<<(NANCH + 255) / 256, 256, 0, stream>>>(keys, hist, state, p * 8);
    scan_kernel<<<1, 256, 0, stream>>>(hist, state, p * 8);
  }
  compact_kernel<<<(NANCH + 255) / 256, 256, 0, stream>>>(keys, state,
                                                          &state[3], selPairs);
  sort_gather_kernel<<<1, 1024, 0, stream>>>(selPairs, boxes, sortedBoxes,
                                             sortedKey);
  nms_mask_kernel<<<dim3(NTILE, (NTILE + 7) / 8), dim3(32, 8), 0, stream>>>(
      sortedBoxes, mask);
  nms_reduce_out_kernel<<<1, 128, 0, stream>>>(sortedKey, sortedBoxes, mask, out);
}